// kernel_generated_3_21887153341323
// MI455X (gfx1250) — compile-verified
//
#include <hip/hip_runtime.h>

// fp32 WMMA fragments (wave32): A,B 16x4 / 4x16 -> 2 VGPRs, C/D 16x16 -> 8 VGPRs
typedef __attribute__((ext_vector_type(2))) float v2f;
typedef __attribute__((ext_vector_type(8))) float v8f;

#define MN  49   // 7*7 spatial
#define KB1 192  // 64 in-ch * 3 (H window)
#define KB2 48   // 16 groups * 3 (W window)

// Fused shift-conv: stage1 (roll+H-window GEMM) -> LDS -> stage2 (W-window GEMM).
// Grid: 2 blocks; block b owns k-rows [b*16, b*16+16) of the stage-1 output,
// which are exactly the o2 values stage 2 needs from that slice.
// K dims are reordered (j-major) so window math hoists out of the WMMA loops.
__global__ __launch_bounds__(256)
void fused_shift_conv_wmma(const float* __restrict__ x,   // (1024, 7, 7) = (i*16+o, h, w)
                           const float* __restrict__ w1,  // (32, 64, 3)
                           const float* __restrict__ w2,  // (32, 16, 3)
                           float* __restrict__ out)       // (1024, 7, 7) = (i*32+o2, m, n)
{
    __shared__ float t4s[16 * 16 * MN];   // [k_local(16)][o(16)][mn(49)] = 49KB

    const int lane  = threadIdx.x & 31;
    const int wave  = threadIdx.x >> 5;   // 0..7
    const int hi    = lane >> 4;          // selects K pair / upper D rows
    const int lo    = lane & 15;          // M row (A) / N col (B, D)
    const int kbase = blockIdx.x * 16;    // stage-1 M rows, stage-2 o2 base

    // ------------- Stage 1: t4[k, o, mn] = sum_{i,j} x_shifted * w1 -------------
    // K reordered as kdim = j*64 + i  (A[row][kdim] = w1[row, i, j])
    for (int job = wave; job < 64; job += 8) {
        const int o  = job >> 2;          // 0..15
        const int nt = job & 3;           // 0..3
        const int c  = nt * 16 + lo;      // tile column (mn index, valid < 49)
        const int m  = c / 7;
        const int n  = c % 7;

        const float* w1row = w1 + (kbase + lo) * KB1;   // this lane's A row

        v8f acc = {};
        #pragma unroll
        for (int jw = 0; jw < 3; ++jw) {                // H-window position (hoisted)
            const int   h    = m + jw - 1;              // pre-roll row
            const bool  ok   = (c < MN) & (h >= 0) & (h < 7);
            const int   xoff = ((h + 6) % 7) * 7 + n;   // rolled row offset (always in-bounds)
            const float* xb  = x + o * MN + xoff;       // element i at xb[i*16*MN]

            #pragma unroll
            for (int k0 = 0; k0 < 64; k0 += 4) {        // 16 WMMA per window pos
                const int i0 = k0 + 2 * hi;
                v2f a, b;
                a.x = w1row[i0 * 3 + jw];
                a.y = w1row[i0 * 3 + jw + 3];
                const float v0 = xb[i0 * (16 * MN)];
                const float v1 = xb[(i0 + 1) * (16 * MN)];
                b.x = ok ? v0 : 0.0f;
                b.y = ok ? v1 : 0.0f;
                acc = __builtin_amdgcn_wmma_f32_16x16x4_f32(false, a, false, b,
                                                            (short)0, acc, false, false);
            }
        }
        if (c < MN) {
            #pragma unroll
            for (int v = 0; v < 8; ++v) {
                const int r = v + 8 * hi;               // local k row
                t4s[(r * 16 + o) * MN + c] = acc[v];
            }
        }
    }

    __syncthreads();

    // ------------- Stage 2: out[i, o2, mn] = sum_{j,kw} t4_window * w2 -------------
    // K reordered as kdim = kw*16 + j  (A[row][kdim] = w2[row, j, kw])
    for (int job = wave; job < 128; job += 8) {
        const int nt  = job & 3;          // 0..3
        const int imt = (job >> 2) & 1;   // 0..1 (i tile)
        const int o2l = job >> 3;         // 0..15 (local o2)
        const int c   = nt * 16 + lo;
        const int m   = c / 7;
        const int n   = c % 7;

        const float* w2row = w2 + (imt * 16 + lo) * KB2;  // this lane's A row

        v8f acc = {};
        #pragma unroll
        for (int kw = 0; kw < 3; ++kw) {                 // W-window position (hoisted)
            const int  w   = n + kw - 1;
            const bool ok  = (c < MN) & (w >= 0) & (w < 7);
            const int  off = ok ? (m * 7 + w) : 0;       // safe LDS column
            const float* tb = t4s + o2l * (16 * MN) + off;  // row j at tb[j*MN]

            #pragma unroll
            for (int k0 = 0; k0 < 16; k0 += 4) {         // 4 WMMA per window pos
                const int j0 = k0 + 2 * hi;
                v2f a, b;
                a.x = w2row[j0 * 3 + kw];
                a.y = w2row[j0 * 3 + kw + 3];
                const float v0 = tb[j0 * MN];
                const float v1 = tb[(j0 + 1) * MN];
                b.x = ok ? v0 : 0.0f;
                b.y = ok ? v1 : 0.0f;
                acc = __builtin_amdgcn_wmma_f32_16x16x4_f32(false, a, false, b,
                                                            (short)0, acc, false, false);
            }
        }
        if (c < MN) {
            #pragma unroll
            for (int v = 0; v < 8; ++v) {
                const int i  = imt * 16 + v + 8 * hi;    // i channel
                const int o2 = kbase + o2l;              // o2 channel
                out[(i * 32 + o2) * MN + c] = acc[v];
            }
        }
    }
}

extern "C" void kernel_launch(void* const* d_in, const int* in_sizes, int n_in,
                              void* d_out, int out_size, void* d_ws, size_t ws_size,
                              hipStream_t stream) {
    const float* x  = (const float*)d_in[0];   // (1,1024,7,7)
    const float* w1 = (const float*)d_in[1];   // (32,64,3)
    const float* w2 = (const float*)d_in[2];   // (32,16,3)
    float* out = (float*)d_out;                // (1,1024,7,7)
    (void)in_sizes; (void)n_in; (void)out_size; (void)d_ws; (void)ws_size;
    fused_shift_conv_wmma<<<dim3(2), dim3(256), 0, stream>>>(x, w1, w2, out);
}